// LayoutRenderModel_22153441313314
// MI455X (gfx1250) — compile-verified
//
#include <hip/hip_runtime.h>
#include <hip/hip_bf16.h>
#include <stdint.h>

#define MAXB 64
#define RB_EPS 0.0001f
#define RB_BIG 10000000000.0f

// Async global->LDS staging (CDNA5 ASYNCcnt path), guarded so the file
// compiles on host pass and on toolchains lacking the builtin.
#if defined(__HIP_DEVICE_COMPILE__) && __has_builtin(__builtin_amdgcn_global_load_async_to_lds_b128)
#define USE_ASYNC_LDS 1
#else
#define USE_ASYNC_LDS 0
#endif

#if USE_ASYNC_LDS
typedef int v4i __attribute__((ext_vector_type(4)));
typedef __attribute__((address_space(1))) v4i gv4i;   // global
typedef __attribute__((address_space(3))) v4i lv4i;   // LDS
#endif

typedef float vf2 __attribute__((ext_vector_type(2)));

static __device__ __forceinline__ vf2 fma2(vf2 a, vf2 b, vf2 c) {
    return __builtin_elementwise_fma(a, b, c);   // -> v_pk_fma_f32
}

__global__ __launch_bounds__(256) void raybox_nearest_kernel(
    const float* __restrict__ rays_o,
    const float* __restrict__ rays_d,
    const float* __restrict__ bbox,
    float* __restrict__ out_idx,   // R floats
    float* __restrict__ out_dist,  // R floats
    int R, int nB)
{
    __shared__ float  raw[MAXB * 6];     // staged raw bbox: center.xyz, size.xyz
    __shared__ float4 bd[MAXB * 3];      // derived per box: 3 x float4 (pk-pair layout)

    const int tid    = threadIdx.x;
    const int chunks = (nB * 6) >> 2;    // 16-byte chunks (B*6 floats, mult of 4)

#if USE_ASYNC_LDS
    if (tid < chunks) {
        gv4i* gsrc = (gv4i*)(uintptr_t)(bbox + tid * 4);
        lv4i* ldst = (lv4i*)(uint32_t)(uintptr_t)(&raw[tid * 4]);
        // b128 async DMA: global -> LDS, tracked by ASYNCcnt
        __builtin_amdgcn_global_load_async_to_lds_b128(gsrc, ldst, 0, 0);
    }
#if __has_builtin(__builtin_amdgcn_s_wait_asynccnt)
    __builtin_amdgcn_s_wait_asynccnt(0);
#else
    asm volatile("s_wait_asynccnt 0" ::: "memory");
#endif
#else
    if (tid < chunks) {
        ((float4*)raw)[tid] = ((const float4*)bbox)[tid];
    }
#endif
    __syncthreads();

    // One thread per box: derive bmin/bmax and EPS-padded bounds once per block.
    // Layout (per box, pk-friendly): axis bounds adjacent in pairs:
    //   f4a = (bminx, bmaxx, bminy, bmaxy)
    //   f4b = (bminz, bmaxz, eminx, eminy)     e* = EPS-padded bounds
    //   f4c = (eminz, emaxx, emaxy, emaxz)
    if (tid < nB) {
        const float cx = raw[tid * 6 + 0], cy = raw[tid * 6 + 1], cz = raw[tid * 6 + 2];
        const float hx = 0.5f * raw[tid * 6 + 3];
        const float hy = 0.5f * raw[tid * 6 + 4];
        const float hz = 0.5f * raw[tid * 6 + 5];
        const float bnx = cx - hx, bny = cy - hy, bnz = cz - hz;
        const float bxx = cx + hx, bxy = cy + hy, bxz = cz + hz;
        bd[tid * 3 + 0] = make_float4(bnx, bxx, bny, bxy);
        bd[tid * 3 + 1] = make_float4(bnz, bxz, bnx - RB_EPS, bny - RB_EPS);
        bd[tid * 3 + 2] = make_float4(bnz - RB_EPS, bxx + RB_EPS, bxy + RB_EPS, bxz + RB_EPS);
    }
    __syncthreads();

    const int i = blockIdx.x * blockDim.x + tid;
    if (i >= R) return;

    const float ox = rays_o[i * 3 + 0];
    const float oy = rays_o[i * 3 + 1];
    const float oz = rays_o[i * 3 + 2];
    float dx = rays_d[i * 3 + 0];
    float dy = rays_d[i * 3 + 1];
    float dz = rays_d[i * 3 + 2];
    dx = (dx == 0.0f) ? 1e-8f : dx;
    dy = (dy == 0.0f) ? 1e-8f : dy;
    dz = (dz == 0.0f) ? 1e-8f : dz;

    const float ivx = 1.0f / dx, ivy = 1.0f / dy, ivz = 1.0f / dz;
    const float nox = -ox * ivx, noy = -oy * ivy, noz = -oz * ivz;

    const vf2 ivx2 = {ivx, ivx}, ivy2 = {ivy, ivy}, ivz2 = {ivz, ivz};
    const vf2 nox2 = {nox, nox}, noy2 = {noy, noy}, noz2 = {noz, noz};
    const vf2 dxy2 = {dx, dy},   oxy2 = {ox, oy};

    float best = RB_BIG;
    int   bi   = 0;
    bool  keep = false;

#pragma unroll 8
    for (int b = 0; b < nB; ++b) {
        const float4 f4a = bd[b * 3 + 0];
        const float4 f4b = bd[b * 3 + 1];
        const float4 f4c = bd[b * 3 + 2];

        // packed slab t's: (bmin,bmax)*inv_d - o*inv_d  -> 3x v_pk_fma_f32
        const vf2 tx2 = fma2((vf2){f4a.x, f4a.y}, ivx2, nox2);
        const vf2 ty2 = fma2((vf2){f4a.z, f4a.w}, ivy2, noy2);
        const vf2 tz2 = fma2((vf2){f4b.x, f4b.y}, ivz2, noz2);

        const float tnx = fminf(tx2.x, tx2.y), txx = fmaxf(tx2.x, tx2.y);
        const float tny = fminf(ty2.x, ty2.y), txy = fmaxf(ty2.x, ty2.y);
        const float tnz = fminf(tz2.x, tz2.y), txz = fmaxf(tz2.x, tz2.y);

        const float tent = fmaxf(fmaxf(tnx, tny), tnz);   // -> v_max3
        const float texi = fminf(fminf(txx, txy), txz);   // -> v_min3

        keep = keep | ((tent < texi) & (texi > 0.0f));

        // entry point must lie inside EPS-padded box, and tent >= 0
        const vf2  tent2 = {tent, tent};
        const vf2  pxy   = fma2(tent2, dxy2, oxy2);       // (px, py) packed
        const float pz   = fmaf(tent, dz, oz);
        const bool valid = (pxy.x >= f4b.z) & (pxy.x <= f4c.y) &
                           (pxy.y >= f4b.w) & (pxy.y <= f4c.z) &
                           (pz    >= f4c.x) & (pz    <= f4c.w) &
                           (tent >= 0.0f);

        const float tn = valid ? tent : RB_BIG;
        if (tn < best) { best = tn; bi = b; }   // strict '<' keeps first argmin
    }

    out_idx[i]  = keep ? (float)bi : -1.0f;
    out_dist[i] = keep ? best : -1.0f;
}

extern "C" void kernel_launch(void* const* d_in, const int* in_sizes, int n_in,
                              void* d_out, int out_size, void* d_ws, size_t ws_size,
                              hipStream_t stream) {
    const float* rays_o = (const float*)d_in[0];
    const float* rays_d = (const float*)d_in[1];
    const float* bbox   = (const float*)d_in[2];
    const int R  = in_sizes[0] / 3;
    int nB = in_sizes[2] / 6;
    if (nB > MAXB) nB = MAXB;

    float* out = (float*)d_out;          // [0,R): idx as float, [R,2R): distance
    const int threads = 256;             // 8 wave32 waves per block
    const int blocks  = (R + threads - 1) / threads;
    hipLaunchKernelGGL(raybox_nearest_kernel, dim3(blocks), dim3(threads), 0, stream,
                       rays_o, rays_d, bbox, out, out + R, R, nB);
}